// GNNModel_69956427317871
// MI455X (gfx1250) — compile-verified
//
#include <hip/hip_runtime.h>
#include <hip/hip_bf16.h>

typedef __attribute__((ext_vector_type(2))) float f32x2;
typedef __attribute__((ext_vector_type(8))) float f32x8;

#define TILE_K 64

// ---------------------------------------------------------------------------
// deg[i] = 1 (self loop)
__global__ void k_fill1(float* __restrict__ deg, int n) {
    int i = blockIdx.x * blockDim.x + threadIdx.x;
    if (i < n) deg[i] = 1.0f;
}

// deg[dst[e]] += 1 for every edge
__global__ void k_degree(const int* __restrict__ dst, float* __restrict__ deg, int E) {
    int e = blockIdx.x * blockDim.x + threadIdx.x;
    if (e < E) atomicAdd(&deg[dst[e]], 1.0f);
}

// dinv[i] = rsqrt(max(deg,1))
__global__ void k_rsqrt(float* __restrict__ deg, int n) {
    int i = blockIdx.x * blockDim.x + threadIdx.x;
    if (i < n) deg[i] = rsqrtf(fmaxf(deg[i], 1.0f));
}

// zero a float buffer
__global__ void k_zero(float* __restrict__ p, long n) {
    long i = (long)blockIdx.x * blockDim.x + threadIdx.x;
    if (i < n) p[i] = 0.0f;
}

// ---------------------------------------------------------------------------
// C[M,N] = A[M,K] @ B[K,N]   (N <= 128), fp32 WMMA 16x16x4.
// Block = 256 threads = 8 waves. blockIdx.x selects a 16-row M tile.
// Wave w computes output columns [16w, 16w+16). A tile shared across waves via
// LDS; B K-chunk (64 x 128) also staged in LDS once per chunk.
__global__ __launch_bounds__(256) void k_gemm_wmma(
    const float* __restrict__ A, const float* __restrict__ B,
    float* __restrict__ C, int M, int K, int N) {
    __shared__ float As[16][TILE_K + 4];
    __shared__ float Bs[TILE_K][128 + 4];

    const int mbase = blockIdx.x * 16;
    const int t    = threadIdx.x;
    const int lane = t & 31;
    const int wave = t >> 5;
    const int col0 = wave * 16;

    f32x8 acc = {};

    const int arow = lane & 15;              // A row within tile / B col within wave slice
    const int kb   = (lane >> 4) << 1;       // K sub-offset: lanes 0-15 -> K+0,1 ; 16-31 -> K+2,3
    const int bcol = col0 + (lane & 15);

    for (int kc = 0; kc < K; kc += TILE_K) {
        // stage A: 16 x TILE_K (zero-fill K remainder)
        for (int i = t; i < 16 * TILE_K; i += 256) {
            int r  = i / TILE_K;
            int kk = i - r * TILE_K;
            int gk = kc + kk;
            int gr = mbase + r;
            As[r][kk] = (gk < K && gr < M) ? A[(size_t)gr * K + gk] : 0.0f;
        }
        // stage B: TILE_K x 128 (zero-fill K and N remainders)
        for (int i = t; i < TILE_K * 128; i += 256) {
            int kk = i >> 7;
            int c  = i & 127;
            int gk = kc + kk;
            Bs[kk][c] = (gk < K && c < N) ? B[(size_t)gk * N + c] : 0.0f;
        }
        __syncthreads();

#pragma unroll
        for (int kk = 0; kk < TILE_K; kk += 4) {
            f32x2 a, b;
            a.x = As[arow][kk + kb];
            a.y = As[arow][kk + kb + 1];
            b.x = Bs[kk + kb][bcol];
            b.y = Bs[kk + kb + 1][bcol];
            acc = __builtin_amdgcn_wmma_f32_16x16x4_f32(
                false, a, false, b, (short)0, acc, false, false);
        }
        __syncthreads();
    }

    // store: VGPR r -> row (mbase + r + (lane>=16 ? 8 : 0)), col = col0 + (lane&15)
    const int col = col0 + (lane & 15);
    if (col < N) {
        const int radd = (lane >> 4) << 3;
#pragma unroll
        for (int r = 0; r < 8; ++r) {
            int gr = mbase + r + radd;
            if (gr < M) C[(size_t)gr * N + col] = acc[r];
        }
    }
}

// ---------------------------------------------------------------------------
// One wave per edge: out[dst] += hw[src] * (dinv[src]*dinv[dst]) over C channels.
__global__ __launch_bounds__(256) void k_scatter(
    const float* __restrict__ hw, const int* __restrict__ src,
    const int* __restrict__ dst, const float* __restrict__ dinv,
    float* __restrict__ out, int E, int C) {
    int gw   = (int)(((long)blockIdx.x * blockDim.x + threadIdx.x) >> 5);
    int lane = threadIdx.x & 31;
    if (gw >= E) return;
    int s = src[gw];
    int d = dst[gw];
    float w = dinv[s] * dinv[d];
    const float* hs = hw + (size_t)s * C;
    float* od = out + (size_t)d * C;
    for (int c = lane; c < C; c += 32)
        atomicAdd(&od[c], hs[c] * w);
}

// ---------------------------------------------------------------------------
// Fused epilogue: agg[i,c] = [relu]( agg[i,c] + hw[i,c]*dinv[i]^2 + bias[c] )
__global__ void k_finish(float* __restrict__ agg, const float* __restrict__ hw,
                         const float* __restrict__ dinv, const float* __restrict__ bias,
                         int n, int C, int do_relu) {
    long i = (long)blockIdx.x * blockDim.x + threadIdx.x;
    if (i >= (long)n * C) return;
    int node = (int)(i / C);
    int c    = (int)(i - (long)node * C);
    float dv = dinv[node];
    float v = agg[i] + hw[i] * dv * dv + bias[c];
    agg[i] = do_relu ? fmaxf(v, 0.0f) : v;
}

// ---------------------------------------------------------------------------
static inline int ceildiv(long a, long b) { return (int)((a + b - 1) / b); }

extern "C" void kernel_launch(void* const* d_in, const int* in_sizes, int n_in,
                              void* d_out, int out_size, void* d_ws, size_t ws_size,
                              hipStream_t stream) {
    const float* x  = (const float*)d_in[0];
    const int*   ei = (const int*)d_in[1];
    const float* W1 = (const float*)d_in[2];
    const float* b1 = (const float*)d_in[3];
    const float* W2 = (const float*)d_in[4];
    const float* b2 = (const float*)d_in[5];
    const float* W3 = (const float*)d_in[6];
    const float* b3 = (const float*)d_in[7];

    const int HID  = in_sizes[3];                 // 128
    const int INC  = in_sizes[2] / HID;           // 1433
    const int N    = in_sizes[0] / INC;           // 50000
    const int OUTC = in_sizes[7];                 // 47
    const int E    = in_sizes[1] / 2;             // 1600000

    const int* src = ei;
    const int* dst = ei + E;

    // workspace layout
    char*  ws   = (char*)d_ws;
    float* dinv = (float*)ws;
    size_t off  = ((size_t)N * 4 + 255) & ~(size_t)255;
    float* bufA = (float*)(ws + off);             // hw  (N x HID)
    float* bufB = bufA + (size_t)N * HID;         // agg / h (N x HID)

    float* outp = (float*)d_out;

    const int TB = 256;

    // --- norm: deg -> dinv
    k_fill1 <<<ceildiv(N, TB), TB, 0, stream>>>(dinv, N);
    k_degree<<<ceildiv(E, TB), TB, 0, stream>>>(dst, dinv, E);
    k_rsqrt <<<ceildiv(N, TB), TB, 0, stream>>>(dinv, N);

    const int mtiles = ceildiv(N, 16);
    const long sc_thr = (long)E * 32;

    // --- layer 1: hw1 = X@W1 -> bufA; agg -> bufB; h1 = relu(...) in bufB
    k_gemm_wmma<<<mtiles, TB, 0, stream>>>(x, W1, bufA, N, INC, HID);
    k_zero     <<<ceildiv((long)N * HID, TB), TB, 0, stream>>>(bufB, (long)N * HID);
    k_scatter  <<<ceildiv(sc_thr, TB), TB, 0, stream>>>(bufA, src, dst, dinv, bufB, E, HID);
    k_finish   <<<ceildiv((long)N * HID, TB), TB, 0, stream>>>(bufB, bufA, dinv, b1, N, HID, 1);

    // --- layer 2: hw2 = h1@W2 -> bufA; agg -> bufB; h2 = relu(...) in bufB
    k_gemm_wmma<<<mtiles, TB, 0, stream>>>(bufB, W2, bufA, N, HID, HID);
    k_zero     <<<ceildiv((long)N * HID, TB), TB, 0, stream>>>(bufB, (long)N * HID);
    k_scatter  <<<ceildiv(sc_thr, TB), TB, 0, stream>>>(bufA, src, dst, dinv, bufB, E, HID);
    k_finish   <<<ceildiv((long)N * HID, TB), TB, 0, stream>>>(bufB, bufA, dinv, b2, N, HID, 1);

    // --- layer 3: hw3 = h2@W3 -> bufA (N x OUTC); agg -> d_out; no relu
    k_gemm_wmma<<<mtiles, TB, 0, stream>>>(bufB, W3, bufA, N, HID, OUTC);
    k_zero     <<<ceildiv((long)N * OUTC, TB), TB, 0, stream>>>(outp, (long)N * OUTC);
    k_scatter  <<<ceildiv(sc_thr, TB), TB, 0, stream>>>(bufA, src, dst, dinv, outp, E, OUTC);
    k_finish   <<<ceildiv((long)N * OUTC, TB), TB, 0, stream>>>(outp, bufA, dinv, b3, N, OUTC, 0);
}